// Group_18305150615660
// MI455X (gfx1250) — compile-verified
//
#include <hip/hip_runtime.h>

// Problem shape from setup_inputs(): xyz (32, 8192, 3) f32, num_group=256, group_size=32.
// Output tuple: patch (32,256,32,3) then center (32,256,3), concatenated flat f32.
constexpr int B = 32;
constexpr int N = 8192;
constexpr int G = 256;
constexpr int K = 32;

constexpr int CHUNK = 512;          // distance columns per LDS chunk (16 x 512 x 4B = 32KB)
constexpr int NW    = 16;           // waves per block in knn kernel (one per center row)
constexpr int BLK2  = NW * 32;      // 512 threads

typedef __attribute__((ext_vector_type(2))) float v2f;
typedef __attribute__((ext_vector_type(8))) float v8f;

// ---------------------------------------------------------------------------
// Kernel 1: Farthest Point Sampling. One block per batch. min_d lives in LDS.
// Matches reference: start index 0; emitted index g is the one selected at the
// *previous* step; argmax ties -> lowest index.
// ---------------------------------------------------------------------------
__global__ __launch_bounds__(1024) void fps_kernel(const float* __restrict__ xyz,
                                                   float* __restrict__ centers) {
  __shared__ float sMin[N];          // 32 KB
  __shared__ float sCur[3];
  __shared__ float sPartV[32];
  __shared__ int   sPartI[32];
  __shared__ int   sNext;

  const int b    = blockIdx.x;
  const int tid  = threadIdx.x;
  const int lane = tid & 31;
  const int wv   = tid >> 5;
  const float* __restrict__ pts = xyz + (size_t)b * N * 3;

  for (int i = tid; i < N; i += 1024) sMin[i] = INFINITY;
  int last = 0;
  __syncthreads();

  for (int g = 0; g < G; ++g) {
    if (tid == 0) {
      float cx = pts[last * 3 + 0], cy = pts[last * 3 + 1], cz = pts[last * 3 + 2];
      sCur[0] = cx; sCur[1] = cy; sCur[2] = cz;
      float* co = centers + ((size_t)b * G + g) * 3;
      co[0] = cx; co[1] = cy; co[2] = cz;
    }
    __syncthreads();
    if (g == G - 1) break;                 // last center written; no update needed

    const float cx = sCur[0], cy = sCur[1], cz = sCur[2];
    float bestv = -INFINITY;
    int   besti = 0;
    for (int i = tid; i < N; i += 1024) {  // 8 strided points per thread
      float dx = pts[i * 3 + 0] - cx;
      float dy = pts[i * 3 + 1] - cy;
      float dz = pts[i * 3 + 2] - cz;
      float d  = dx * dx + dy * dy + dz * dz;
      float m  = fminf(sMin[i], d);
      sMin[i]  = m;
      if (m > bestv || (m == bestv && i < besti)) { bestv = m; besti = i; }
    }
    // wave32 reduction (argmax, tie -> smaller index)
    #pragma unroll
    for (int off = 16; off > 0; off >>= 1) {
      float ov = __shfl_down(bestv, off, 32);
      int   oi = __shfl_down(besti, off, 32);
      if (ov > bestv || (ov == bestv && oi < besti)) { bestv = ov; besti = oi; }
    }
    if (lane == 0) { sPartV[wv] = bestv; sPartI[wv] = besti; }
    __syncthreads();
    if (wv == 0) {
      float v = sPartV[lane];
      int   i = sPartI[lane];
      #pragma unroll
      for (int off = 16; off > 0; off >>= 1) {
        float ov = __shfl_down(v, off, 32);
        int   oi = __shfl_down(i, off, 32);
        if (ov > v || (ov == v && oi < i)) { v = ov; i = oi; }
      }
      if (lane == 0) sNext = i;
    }
    __syncthreads();
    last = sNext;
  }
}

// ---------------------------------------------------------------------------
// Kernel 2: fused WMMA distance tiles + wave-parallel sorted top-32 + gather.
// Block = (batch b, 16 centers). 16 waves: compute phase fills a 16xCHUNK
// squared-distance slab in LDS (one v_wmma_f32_16x16x4_f32 per 16x16 tile,
// accumulator pre-seeded with |c|^2); select phase: wave w keeps a sorted
// 32-entry list distributed one entry per lane, ballot-pruned insertion.
// B-operand construction is branchless (cndmask selects) so the half-wave
// K-split does not create exec-mask branches in the hot loop.
// ---------------------------------------------------------------------------
__global__ __launch_bounds__(BLK2) void knn_group_kernel(const float* __restrict__ xyz,
                                                         const float* __restrict__ centers,
                                                         float* __restrict__ patch) {
  __shared__ float sD[16 * CHUNK];   // 32 KB distance slab
  __shared__ float sC[16][3];
  __shared__ float sN2[16];

  const int b    = blockIdx.x / (G / 16);
  const int cg   = blockIdx.x % (G / 16);
  const int g0   = cg * 16;
  const int tid  = threadIdx.x;
  const int lane = tid & 31;
  const int wv   = tid >> 5;         // 0..15, owns center row wv
  const int hm   = lane & 15;        // row/col within half-wave
  const bool hi  = lane >= 16;

  const float* __restrict__ pts = xyz + (size_t)b * N * 3;

  if (tid < 16) {
    const float* c = centers + ((size_t)b * G + g0 + tid) * 3;
    float cx = c[0], cy = c[1], cz = c[2];
    sC[tid][0] = cx; sC[tid][1] = cy; sC[tid][2] = cz;
    sN2[tid]   = cx * cx + cy * cy + cz * cz;
  }
  __syncthreads();

  // A operand (16x4 f32): lanes 0-15 hold row m, K=0..1; lanes 16-31 row m, K=2..3.
  // Row m = (-2cx, -2cy, -2cz, 1).  Branchless selects.
  const float acx = sC[hm][0], acy = sC[hm][1], acz = sC[hm][2];
  v2f Aop;
  Aop.x = hi ? (-2.0f * acz) : (-2.0f * acx);
  Aop.y = hi ? 1.0f          : (-2.0f * acy);
  // C accumulator seed: C[m][n] = |c_m|^2  (VGPR r: lanes0-15 -> m=r, lanes16-31 -> m=r+8)
  v8f Cseed;
  #pragma unroll
  for (int r = 0; r < 8; ++r) Cseed[r] = sN2[r + (hi ? 8 : 0)];

  // distributed sorted top-K: lane i holds i-th smallest (dist, idx)
  float e_d = INFINITY;
  int   e_i = 0x7fffffff;
  float worst = INFINITY;

  for (int ch = 0; ch < N / CHUNK; ++ch) {
    const int cbase = ch * CHUNK;
    // ----- compute phase: 32 tiles per chunk, 2 per wave -----
    #pragma unroll
    for (int t = 0; t < 2; ++t) {
      const int tile = 2 * wv + t;
      const int n0   = cbase + tile * 16;
      const int j    = n0 + hm;                       // point index for this lane
      if (t == 0 && ch + 1 < N / CHUNK)               // prefetch next chunk's points
        __builtin_prefetch(&pts[(size_t)(cbase + CHUNK + tile * 16 + hm) * 3], 0, 1);
      // Unconditional 3-component load (both half-waves address the same point),
      // then branchless K-split selects -> single b96 load, no exec branches.
      const float px = pts[j * 3 + 0];
      const float py = pts[j * 3 + 1];
      const float pz = pts[j * 3 + 2];
      const float n2 = px * px + py * py + pz * pz;
      // B operand (4x16 f32): col n = j. Col = (x, y, z, |p|^2).
      // VGPR0: lanes0-15 K=0, lanes16-31 K=2; VGPR1: lanes0-15 K=1, lanes16-31 K=3.
      v2f Bop;
      Bop.x = hi ? pz : px;
      Bop.y = hi ? n2 : py;
      // D[m][n] = |c|^2 + |p|^2 - 2 c.p   in ONE wmma
      v8f D = __builtin_amdgcn_wmma_f32_16x16x4_f32(
          /*neg_a=*/false, Aop, /*neg_b=*/false, Bop,
          /*c_mod=*/(short)0, Cseed, /*reuse_a=*/false, /*reuse_b=*/false);
      #pragma unroll
      for (int r = 0; r < 8; ++r) {
        const int row = r + (hi ? 8 : 0);
        sD[row * CHUNK + tile * 16 + hm] = D[r];
      }
    }
    __syncthreads();

    // ----- select phase: wave wv scans row wv of the slab -----
    for (int it = 0; it < CHUNK / 32; ++it) {
      const int col = it * 32 + lane;
      const float d = sD[wv * CHUNK + col];
      const int   j = cbase + col;
      unsigned long long m = __ballot(d < worst);
      while (m) {                                    // rare after warm-up
        const int s = __ffsll(m) - 1;
        m &= m - 1;
        const float dd = __shfl(d, s, 32);
        const int   jj = __shfl(j, s, 32);
        if (dd < worst) {
          // stable insertion position: after all entries with (e_d <= dd)
          const int pos = __popcll(__ballot(e_d <= dd));
          if (pos < 32) {
            const float sd = __shfl_up(e_d, 1, 32);
            const int   si = __shfl_up(e_i, 1, 32);
            if (lane > pos)       { e_d = sd; e_i = si; }
            else if (lane == pos) { e_d = dd; e_i = jj; }
            worst = __shfl(e_d, 31, 32);
          }
        }
      }
    }
    __syncthreads();                                 // slab reused next chunk
  }

  // ----- gather + recenter: lane k writes patch[b, g0+wv, k, :] -----
  const int g = g0 + wv;
  const float cx = sC[wv][0], cy = sC[wv][1], cz = sC[wv][2];
  float* o = patch + (((size_t)b * G + g) * K + lane) * 3;
  o[0] = pts[e_i * 3 + 0] - cx;
  o[1] = pts[e_i * 3 + 1] - cy;
  o[2] = pts[e_i * 3 + 2] - cz;
}

extern "C" void kernel_launch(void* const* d_in, const int* in_sizes, int n_in,
                              void* d_out, int out_size, void* d_ws, size_t ws_size,
                              hipStream_t stream) {
  (void)in_sizes; (void)n_in; (void)d_ws; (void)ws_size; (void)out_size;
  const float* xyz = (const float*)d_in[0];
  float* patch   = (float*)d_out;                 // (32,256,32,3)
  float* centers = patch + (size_t)B * G * K * 3; // (32,256,3)

  fps_kernel<<<B, 1024, 0, stream>>>(xyz, centers);
  knn_group_kernel<<<B * (G / 16), BLK2, 0, stream>>>(xyz, centers, patch);
}